// PredictionNetwork_57183194579285
// MI455X (gfx1250) — compile-verified
//
#include <hip/hip_runtime.h>

typedef __attribute__((ext_vector_type(2))) float v2f;
typedef __attribute__((ext_vector_type(4))) float v4f;
typedef __attribute__((ext_vector_type(8))) float v8f;

// ---------------------------------------------------------------------------
// Stage 1: c_proj[b,s,o] = sum_d c[b,s,d] * W[k][o][d] + bias[k][o]
// GEMM: M = B*S = 2048 (rows of c), N = 512 (o), K = 512 (d)
// One wave per 16x16 output tile, K-loop with V_WMMA_F32_16X16X4_F32.
// ---------------------------------------------------------------------------
__global__ __launch_bounds__(256) void cproj_gemm_wmma(
    const float* __restrict__ c,      // [2048, 512]
    const float* __restrict__ W,      // [12, 512, 512]
    const float* __restrict__ bias,   // [12, 512]
    const int*   __restrict__ kptr,
    float*       __restrict__ cproj)  // [2048, 512]
{
    const int D = 512;
    const int k = kptr[0];
    const float* __restrict__ Wk = W    + (size_t)k * D * D;
    const float* __restrict__ bk = bias + (size_t)k * D;

    const int wave   = (blockIdx.x * blockDim.x + threadIdx.x) >> 5; // 0..4095
    const int lane   = threadIdx.x & 31;
    const int tile_m = wave >> 5;   // 0..127  (rows of c, 16 each)
    const int tile_n = wave & 31;   // 0..31   (output dims, 16 each)
    const int half   = lane >> 4;   // 0/1 : lanes 16-31 hold K+2..K+3
    const int lm     = lane & 15;

    // A fragment source: row M = tile_m*16 + lm of c, K pair = k0 + 2*half
    const float* __restrict__ arow = c  + (size_t)(tile_m * 16 + lm) * D + 2 * half;
    // B fragment source: B[k][n] = Wk[n][k] (row-major Wk rows are contiguous in d)
    const float* __restrict__ brow = Wk + (size_t)(tile_n * 16 + lm) * D + 2 * half;

    v8f acc = {};
#pragma unroll 8
    for (int k0 = 0; k0 < D; k0 += 4) {
        v2f a = *(const v2f*)(arow + k0);
        v2f b = *(const v2f*)(brow + k0);
        // (neg_a, A, neg_b, B, c_mod, C, reuse_a, reuse_b)
        acc = __builtin_amdgcn_wmma_f32_16x16x4_f32(
            false, a, false, b, (short)0, acc, false, false);
    }

    // C/D layout: VGPR r holds M = r + 8*half, column N = lm (fixed per lane)
    const float bv = bk[tile_n * 16 + lm];
    float* __restrict__ out =
        cproj + (size_t)(tile_m * 16 + 8 * half) * D + tile_n * 16 + lm;
#pragma unroll
    for (int r = 0; r < 8; ++r)
        out[(size_t)r * D] = acc[r] + bv;
}

// ---------------------------------------------------------------------------
// Stage 2: out[b,n,s] = dot(c_proj[b,s,:], z[b,n,s,:]) / 512
// Pure HBM stream over z (536 MB). One wave -> 4 outputs sharing the same
// (b,s): c_proj chunk kept in registers, z loaded nontemporally (no reuse),
// wave32 xor-shuffle reduction.
// ---------------------------------------------------------------------------
__global__ __launch_bounds__(256) void cpc_dot_stream(
    const float* __restrict__ z,      // [16,128,128,512]
    const float* __restrict__ cproj,  // [2048, 512]
    float*       __restrict__ out)    // [16,128,128]
{
    const int D = 512;
    const int wave = (blockIdx.x * blockDim.x + threadIdx.x) >> 5; // 0..65535
    const int lane = threadIdx.x & 31;

    // wave -> (b, ngroup, s) ; 4 consecutive n per wave
    const int s  = wave & 127;
    const int t  = wave >> 7;
    const int ng = t & 31;
    const int b  = t >> 5;
    const int n0 = ng * 4;

    // lane's 16-float slice of c_proj[b,s,:] (L2-resident, reused 128x)
    const float* __restrict__ cp = cproj + ((size_t)(b * 128 + s)) * D + lane * 16;
    const v4f c0 = *(const v4f*)(cp + 0);
    const v4f c1 = *(const v4f*)(cp + 4);
    const v4f c2 = *(const v4f*)(cp + 8);
    const v4f c3 = *(const v4f*)(cp + 12);

    float sums[4];
#pragma unroll
    for (int j = 0; j < 4; ++j) {
        const int n = n0 + j;
        const size_t zrow = (((size_t)b * 128 + n) * 128 + s) * (size_t)D;
        const v4f* __restrict__ zp = (const v4f*)(z + zrow + lane * 16);
        v4f z0 = __builtin_nontemporal_load(zp + 0);
        v4f z1 = __builtin_nontemporal_load(zp + 1);
        v4f z2 = __builtin_nontemporal_load(zp + 2);
        v4f z3 = __builtin_nontemporal_load(zp + 3);
        v4f p = z0 * c0;
        p += z1 * c1;
        p += z2 * c2;
        p += z3 * c3;
        float sum = p[0] + p[1] + p[2] + p[3];
#pragma unroll
        for (int off = 16; off > 0; off >>= 1)
            sum += __shfl_xor(sum, off, 32);
        sums[j] = sum;
    }

    if (lane == 0) {
        float* __restrict__ o = out + ((size_t)(b * 128 + n0) * 128 + s);
#pragma unroll
        for (int j = 0; j < 4; ++j)
            o[(size_t)j * 128] = sums[j] * (1.0f / 512.0f);
    }
}

// ---------------------------------------------------------------------------
// Inputs (setup_inputs order): c [16,128,512] f32, z [16,128,128,512] f32,
// W [12,512,512] f32, b [12,512] f32, k int scalar.
// Output: [16,128,128] f32. Workspace: c_proj needs 4 MB.
// ---------------------------------------------------------------------------
extern "C" void kernel_launch(void* const* d_in, const int* in_sizes, int n_in,
                              void* d_out, int out_size, void* d_ws, size_t ws_size,
                              hipStream_t stream) {
    const float* c    = (const float*)d_in[0];
    const float* z    = (const float*)d_in[1];
    const float* W    = (const float*)d_in[2];
    const float* bias = (const float*)d_in[3];
    const int*   k    = (const int*)d_in[4];
    float* cproj = (float*)d_ws;          // 2048*512 f32 = 4 MB
    float* out   = (float*)d_out;

    // 4096 tiles, 8 waves (256 threads) per block -> 512 blocks
    cproj_gemm_wmma<<<512, 256, 0, stream>>>(c, W, bias, k, cproj);
    // 65536 waves, 8 per block -> 8192 blocks
    cpc_dot_stream<<<8192, 256, 0, stream>>>(z, cproj, out);
}